// GaussianSample_76020921139870
// MI455X (gfx1250) — compile-verified
//
#include <hip/hip_runtime.h>
#include <hip/hip_bf16.h>

typedef __attribute__((ext_vector_type(2))) float v2f;
typedef __attribute__((ext_vector_type(8))) float v8f;

#define N_   16
#define C_   64
#define T_   30
#define H_   64
#define W_   44
#define OUTW 22
#define ATTH 32
#define OUTH 48
#define PADT 8

// workspace layout (floats)
#define OFF_FXT   0                         // [N][22 kpairs][32 cols][2]  B-layout, cols>=22 zero
#define SZ_FXT    (N_ * 44 * 32)
#define OFF_FY    (OFF_FXT + SZ_FXT)        // [N][32][64]  A-layout
#define SZ_FY     (N_ * 32 * 64)
#define OFF_FT    (OFF_FY + SZ_FY)          // [N][32][32]  A-layout, zero padded
#define SZ_FT     (N_ * 32 * 32)
#define OFF_GAM   (OFF_FT + SZ_FT)          // [N]
#define SZ_GAM    (N_)
#define OFF_G2    (OFF_GAM + SZ_GAM)        // [N][C][T][32][22] packed
#define SZ_G2     ((size_t)N_ * C_ * T_ * ATTH * OUTW)

__device__ __forceinline__ v8f wmma_f32(v2f a, v2f b, v8f c) {
    return __builtin_amdgcn_wmma_f32_16x16x4_f32(false, a, false, b, (short)0, c,
                                                 false, false);
}

// ---------------------------------------------------------------------------
// Kernel 1: filterbanks (tiny). One block per n, 128 threads.
// FxT is stored K-pair interleaved: elem(w, a) -> [(w>>1)*64 + a*2 + (w&1)]
// so stage-1 B fragments are single aligned ds_load_b64s.
// ---------------------------------------------------------------------------
__global__ void fb_kernel(const float* __restrict__ dt, const float* __restrict__ dx,
                          const float* __restrict__ dy, const float* __restrict__ ls2,
                          const float* __restrict__ ldt, const float* __restrict__ ld,
                          const float* __restrict__ lg, float* __restrict__ ws) {
    const int n = blockIdx.x;
    const int tid = threadIdx.x;

    const float sigma2  = expf(ls2[n]);                 // SIGMA_OFF = 1
    const float inv2s   = 1.0f / (2.0f * sigma2);
    const float dt_     = tanhf(dt[n]) * 15.0f + 15.0f; // T/2 + anchor_t
    const float dx_     = tanhf(dx[n]) * 22.0f + 22.0f; // IN_W/2 + anchor_x
    const float dy_     = tanhf(dy[n]) * 32.0f + 32.0f; // IN_H/2 + anchor_y
    const float delta_t = expf(ldt[n]);
    const float delta   = expf(ld[n]);

    float* fxT = ws + OFF_FXT + n * (44 * 32);
    float* fy  = ws + OFF_FY  + n * (32 * 64);
    float* ft  = ws + OFF_FT  + n * (32 * 32);

    if (tid == 0) ws[OFF_GAM + n] = 1.0f / (1.0f + expf(-lg[n]));

    // Fy rows (b = 0..31), A-layout [32][64]
    if (tid < 32) {
        const float mu = dy_ + ((float)tid - 16.0f) * delta;
        float s = 0.0f;
        for (int h = 0; h < H_; ++h) {
            float d = (float)h - mu;
            s += expf(-d * d * inv2s);
        }
        const float r = 1.0f / fmaxf(s, 1e-8f);
        for (int h = 0; h < H_; ++h) {
            float d = (float)h - mu;
            fy[tid * 64 + h] = expf(-d * d * inv2s) * r;
        }
    }
    // Ft rows (s = 0..31), A-layout [32][32] zero-padded
    if (tid >= 32 && tid < 64) {
        const int i = tid - 32;
        if (i < T_) {
            const float mu = dt_ + ((float)i - 15.0f) * delta_t;
            float s = 0.0f;
            for (int g = 0; g < T_; ++g) {
                float d = (float)g - mu;
                s += expf(-d * d * inv2s);
            }
            const float r = 1.0f / fmaxf(s, 1e-8f);
            for (int g = 0; g < 32; ++g) {
                float v = 0.0f;
                if (g < T_) {
                    float d = (float)g - mu;
                    v = expf(-d * d * inv2s) * r;
                }
                ft[i * 32 + g] = v;
            }
        } else {
            for (int g = 0; g < 32; ++g) ft[i * 32 + g] = 0.0f;
        }
    }
    // FxT zero init (covers padding cols 22..31 and all pairs)
    if (tid >= 64 && tid < 64 + 44) {
        const int w = tid - 64;
        for (int a = 0; a < 32; ++a) fxT[(w >> 1) * 64 + a * 2 + (w & 1)] = 0.0f;
    }
    __syncthreads();
    // Fx rows (a = 0..21) scattered into interleaved FxT
    if (tid < OUTW) {
        const float mu = dx_ + ((float)tid - 11.0f) * delta;
        float s = 0.0f;
        for (int w = 0; w < W_; ++w) {
            float d = (float)w - mu;
            s += expf(-d * d * inv2s);
        }
        const float r = 1.0f / fmaxf(s, 1e-8f);
        for (int w = 0; w < W_; ++w) {
            float d = (float)w - mu;
            fxT[(w >> 1) * 64 + tid * 2 + (w & 1)] = expf(-d * d * inv2s) * r;
        }
    }
}

// ---------------------------------------------------------------------------
// Kernel 2: fused stages 1+2 per (n,c,t). One wave (32 lanes) per block.
//   G1[64][32] = X[64][44] * FxT[44][32]      (contract w)
//   G2[32][22] = Fy[32][64] * G1[64][32]      (contract h)
// sFxT and sG1 use K-pair interleaved layout: (k,col) -> [(k>>1)*64 + col*2 + (k&1)]
// ---------------------------------------------------------------------------
__global__ __launch_bounds__(32) void stage12_kernel(const float* __restrict__ x,
                                                     const float* __restrict__ ws,
                                                     float* __restrict__ g2) {
    __shared__ float sFxT[44 * 32];   // interleaved
    __shared__ float sFy[32 * 64];    // row-major A layout
    __shared__ float sG1[64 * 32];    // interleaved

    const int blk = blockIdx.x;
    const int n = blk / (C_ * T_);
    const int rem = blk % (C_ * T_);
    const int c = rem / T_;
    const int t = rem % T_;

    const int lane = threadIdx.x;
    const int lhalf = lane >> 4;   // 0/1: which K pair
    const int llow = lane & 15;

    const float* fxT = ws + OFF_FXT + n * (44 * 32);
    const float* fy  = ws + OFF_FY  + n * (32 * 64);
    for (int i = lane; i < 44 * 32; i += 32) sFxT[i] = fxT[i];
    for (int i = lane; i < 32 * 64; i += 32) sFy[i] = fy[i];
    __syncthreads();

    const float* xp = x + (((size_t)(n * C_ + c) * T_ + t) * H_ * W_);

    // ---- stage 1: G1 = X * FxT ----
    #pragma unroll
    for (int mt = 0; mt < 4; ++mt) {
        v8f acc0 = {};
        v8f acc1 = {};
        const int row = mt * 16 + llow;
        #pragma unroll
        for (int k = 0; k < 11; ++k) {
            const int kk = k * 4 + lhalf * 2;
            const v2f a  = *(const v2f*)&xp[row * W_ + kk];
            const v2f b0 = *(const v2f*)&sFxT[(kk >> 1) * 64 + llow * 2];
            const v2f b1 = *(const v2f*)&sFxT[(kk >> 1) * 64 + (llow + 16) * 2];
            acc0 = wmma_f32(a, b0, acc0);
            acc1 = wmma_f32(a, b1, acc1);
        }
        // rows (r, r+1) form one interleaved pair -> b64 stores
        #pragma unroll
        for (int i = 0; i < 8; i += 2) {
            const int r = mt * 16 + i + lhalf * 8;  // even
            v2f p0; p0.x = acc0[i]; p0.y = acc0[i + 1];
            v2f p1; p1.x = acc1[i]; p1.y = acc1[i + 1];
            *(v2f*)&sG1[(r >> 1) * 64 + llow * 2] = p0;
            *(v2f*)&sG1[(r >> 1) * 64 + (llow + 16) * 2] = p1;
        }
    }
    __syncthreads();

    // ---- stage 2: G2 = Fy * G1 ----
    float* g2p = g2 + ((size_t)(n * C_ + c) * T_ + t) * (ATTH * OUTW);
    #pragma unroll
    for (int mt = 0; mt < 2; ++mt) {
        v8f acc0 = {};
        v8f acc1 = {};
        const int row = mt * 16 + llow;  // b index
        #pragma unroll
        for (int k = 0; k < 16; ++k) {
            const int kk = k * 4 + lhalf * 2;
            const v2f a  = *(const v2f*)&sFy[row * 64 + kk];
            const v2f b0 = *(const v2f*)&sG1[(kk >> 1) * 64 + llow * 2];
            const v2f b1 = *(const v2f*)&sG1[(kk >> 1) * 64 + (llow + 16) * 2];
            acc0 = wmma_f32(a, b0, acc0);
            acc1 = wmma_f32(a, b1, acc1);
        }
        #pragma unroll
        for (int i = 0; i < 8; ++i) {
            const int bb = mt * 16 + i + lhalf * 8;
            g2p[bb * OUTW + llow] = acc0[i];        // cols 0..15
            const int colh = 16 + llow;
            if (colh < OUTW) g2p[bb * OUTW + colh] = acc1[i];  // cols 16..21
        }
    }
}

// ---------------------------------------------------------------------------
// Kernel 3: stage 3 + gamma + H-pad per (n,c). Two waves per block; each wave
// owns half of the 44 n-tiles (wave-uniform control flow around WMMA).
//   Out[s][b*22+a] = sum_t Ft[s][t] * G2[t][b*22+a]   (M=32pad, K=32pad, N=704)
// ---------------------------------------------------------------------------
__global__ __launch_bounds__(64) void stage3_kernel(const float* __restrict__ g2,
                                                    const float* __restrict__ ws,
                                                    float* __restrict__ out) {
    __shared__ float sFt[32 * 32];

    const int n = blockIdx.x / C_;
    const int c = blockIdx.x % C_;
    const int tid = threadIdx.x;
    const int wave = tid >> 5;     // 0/1: n-tile half
    const int lane = tid & 31;
    const int lhalf = lane >> 4;
    const int llow = lane & 15;

    const float* ftp = ws + OFF_FT + n * (32 * 32);
    for (int i = tid; i < 32 * 32; i += 64) sFt[i] = ftp[i];

    float* op = out + (size_t)(n * C_ + c) * T_ * OUTH * OUTW;

    // zero the H-padding rows (rows 0..7 and 40..47 for each s)
    for (int idx = tid; idx < T_ * 2 * PADT * OUTW; idx += 64) {
        const int s = idx / (2 * PADT * OUTW);
        const int rr = (idx / OUTW) % (2 * PADT);
        const int a = idx % OUTW;
        const int r = (rr < PADT) ? rr : (rr + ATTH);  // 0..7 | 40..47
        op[((size_t)s * OUTH + r) * OUTW + a] = 0.0f;
    }

    const float gam = ws[OFF_GAM + n];
    const float* g2p = g2 + (size_t)(n * C_ + c) * T_ * (ATTH * OUTW);  // [t][704]
    __syncthreads();

    #pragma unroll
    for (int mt = 0; mt < 2; ++mt) {
        // preload A fragments (Ft, reused across all n-tiles of this wave)
        v2f af[8];
        const int arow = mt * 16 + llow;  // s
        #pragma unroll
        for (int k = 0; k < 8; ++k) {
            const int kk = k * 4 + lhalf * 2;
            af[k] = *(const v2f*)&sFt[arow * 32 + kk];
        }
        const int nt0 = wave * 22;
        for (int nt = nt0; nt < nt0 + 22; ++nt) {
            v8f acc = {};
            const int col = nt * 16 + llow;  // 0..703
            #pragma unroll
            for (int k = 0; k < 8; ++k) {
                const int kk = k * 4 + lhalf * 2;
                v2f b;
                b.x = (kk < T_) ? g2p[(size_t)kk * 704 + col] : 0.0f;
                b.y = (kk + 1 < T_) ? g2p[(size_t)(kk + 1) * 704 + col] : 0.0f;
                acc = wmma_f32(af[k], b, acc);
            }
            const int bb = col / OUTW;
            const int a = col - bb * OUTW;
            #pragma unroll
            for (int i = 0; i < 8; ++i) {
                const int s = mt * 16 + i + lhalf * 8;
                if (s < T_)
                    op[((size_t)s * OUTH + PADT + bb) * OUTW + a] = gam * acc[i];
            }
        }
    }
}

// ---------------------------------------------------------------------------
extern "C" void kernel_launch(void* const* d_in, const int* in_sizes, int n_in,
                              void* d_out, int out_size, void* d_ws, size_t ws_size,
                              hipStream_t stream) {
    const float* x   = (const float*)d_in[0];
    const float* dt  = (const float*)d_in[1];
    const float* dx  = (const float*)d_in[2];
    const float* dy  = (const float*)d_in[3];
    const float* ls2 = (const float*)d_in[4];
    const float* ldt = (const float*)d_in[5];
    const float* ld  = (const float*)d_in[6];
    const float* lg  = (const float*)d_in[7];

    float* ws = (float*)d_ws;
    float* g2 = ws + OFF_G2;
    float* out = (float*)d_out;

    fb_kernel<<<N_, 128, 0, stream>>>(dt, dx, dy, ls2, ldt, ld, lg, ws);
    stage12_kernel<<<N_ * C_ * T_, 32, 0, stream>>>(x, ws, g2);
    stage3_kernel<<<N_ * C_, 64, 0, stream>>>(g2, ws, out);
}